// GAT_4045859193304
// MI455X (gfx1250) — compile-verified
//
#include <hip/hip_runtime.h>
#include <math.h>

typedef __attribute__((ext_vector_type(2))) float v2f;
typedef __attribute__((ext_vector_type(8))) float v8f;

#define NEG_SLOPE 0.2f

// ---------------- utility ----------------
__global__ void fill_kernel(float* __restrict__ p, long long n, float v) {
    long long i = (long long)blockIdx.x * blockDim.x + threadIdx.x;
    long long stride = (long long)gridDim.x * blockDim.x;
    for (; i < n; i += stride) p[i] = v;
}

__device__ inline void atomicMaxFloat(float* addr, float val) {
    // sign-split bit trick: positives via signed-int max, negatives via unsigned min.
    if (val >= 0.0f) {
        atomicMax((int*)addr, __float_as_int(val));
    } else {
        atomicMin((unsigned int*)addr, (unsigned int)__float_as_uint(val));
    }
}

// ---------------- WMMA f32 GEMM: C[M,Nc] = A[M,K] * B[K,Nc] ----------------
// One wave per 16x16 output tile, K stepped by 4 (V_WMMA_F32_16X16X4_F32).
// A frag per ISA table: lane L<16 -> row M=L holds K=k,k+1; lane>=16 holds K=k+2,k+3.
// B frag (K x 16): lane L<16 -> col N=L, rows K=k (v0), k+1 (v1); lanes>=16 rows k+2,k+3.
// C/D: VGPR r -> M = r + (lane>=16 ? 8 : 0), N = lane&15.
__global__ void wmma_gemm_f32(const float* __restrict__ A, const float* __restrict__ B,
                              float* __restrict__ C, int M, int K, int Nc) {
    int lane = threadIdx.x & 31;
    int wave = threadIdx.x >> 5;
    int tilesN = Nc >> 4;
    int tilesM = M >> 4;              // M is a multiple of 16 for this workload
    int tile = blockIdx.x * (blockDim.x >> 5) + wave;
    if (tile >= tilesM * tilesN) return;   // wave-uniform branch, EXEC stays full
    int tm = tile / tilesN, tn = tile % tilesN;
    int m0 = tm << 4, n0 = tn << 4;
    int row = lane & 15;
    int hi  = lane >> 4;              // 0 or 1

    v8f acc = {};
    for (int k = 0; k < K; k += 4) {
        int ka = k + hi * 2;
        const float* ap = A + (long long)(m0 + row) * K + ka;
        v2f a; a[0] = ap[0]; a[1] = ap[1];
        const float* bp = B + (long long)ka * Nc + n0 + row;
        v2f b; b[0] = bp[0]; b[1] = bp[Nc];
        acc = __builtin_amdgcn_wmma_f32_16x16x4_f32(false, a, false, b,
                                                    (short)0, acc, false, false);
    }
    int crow = m0 + (hi ? 8 : 0);
    int ccol = n0 + row;
    for (int r = 0; r < 8; ++r)
        C[(long long)(crow + r) * Nc + ccol] = acc[r];
}

// ---------------- attention coefficients: alpha_s/alpha_d per (node, head) ----------------
__global__ void alpha_kernel(const float* __restrict__ h, const float* __restrict__ asrc,
                             const float* __restrict__ adst, float* __restrict__ out_s,
                             float* __restrict__ out_d, int N, int H, int C) {
    int idx = blockIdx.x * blockDim.x + threadIdx.x;
    if (idx >= N * H) return;
    int n = idx / H, hh = idx % H;
    const float* hp = h + (long long)n * H * C + (long long)hh * C;
    const float* as = asrc + hh * C;
    const float* ad = adst + hh * C;
    float ss = 0.f, sd = 0.f;
    for (int c = 0; c < C; ++c) { float v = hp[c]; ss += v * as[c]; sd += v * ad[c]; }
    out_s[idx] = ss;
    out_d[idx] = sd;
}

// ---------------- edge softmax pass 1: segment max over dst ----------------
__global__ void edge_max_kernel(const int* __restrict__ ei, const float* __restrict__ as,
                                const float* __restrict__ ad, float* __restrict__ nmax,
                                int E, int ET, int H) {
    int e = blockIdx.x * blockDim.x + threadIdx.x;
    if (e >= ET) return;
    int s = (e < E) ? ei[e]     : (e - E);
    int d = (e < E) ? ei[E + e] : (e - E);
    for (int hh = 0; hh < H; ++hh) {
        float v = as[s * H + hh] + ad[d * H + hh];
        v = v > 0.f ? v : NEG_SLOPE * v;
        atomicMaxFloat(&nmax[d * H + hh], v);
    }
}

// ---------------- edge softmax pass 2: exp + segment sum, stash per-edge weight ----------------
__global__ void edge_expsum_kernel(const int* __restrict__ ei, const float* __restrict__ as,
                                   const float* __restrict__ ad, const float* __restrict__ nmax,
                                   float* __restrict__ nsum, float* __restrict__ ew,
                                   int E, int ET, int H) {
    int e = blockIdx.x * blockDim.x + threadIdx.x;
    if (e >= ET) return;
    int s = (e < E) ? ei[e]     : (e - E);
    int d = (e < E) ? ei[E + e] : (e - E);
    for (int hh = 0; hh < H; ++hh) {
        float v = as[s * H + hh] + ad[d * H + hh];
        v = v > 0.f ? v : NEG_SLOPE * v;
        float w = expf(v - nmax[d * H + hh]);
        ew[(long long)e * H + hh] = w;
        atomicAdd(&nsum[d * H + hh], w);
    }
}

// ---------------- edge scatter: out[dst] += alpha * h[src], one wave per edge ----------------
__global__ void edge_scatter_kernel(const int* __restrict__ ei, const float* __restrict__ hin,
                                    const float* __restrict__ ew, const float* __restrict__ nsum,
                                    float* __restrict__ out, int E, int ET, int H, int C) {
    int lane = threadIdx.x & 31;
    int wave = threadIdx.x >> 5;
    long long e = (long long)blockIdx.x * (blockDim.x >> 5) + wave;
    if (e >= ET) return;                        // wave-uniform
    int s = (e < E) ? ei[e]     : (int)(e - E);
    int d = (e < E) ? ei[E + e] : (int)(e - E);
    int HC = H * C;
    const float* hs = hin + (long long)s * HC;
    float* od = out + (long long)d * HC;
    for (int c = lane; c < HC; c += 32) {
        int hh = c / C;
        float alpha = ew[e * H + hh] / nsum[d * H + hh];
        atomicAdd(&od[c], alpha * hs[c]);
    }
}

// ---------------- bias + ELU epilogue ----------------
__global__ void bias_elu_kernel(const float* __restrict__ acc, const float* __restrict__ bias,
                                float* __restrict__ dst, long long total, int HC) {
    long long i = (long long)blockIdx.x * blockDim.x + threadIdx.x;
    long long stride = (long long)gridDim.x * blockDim.x;
    for (; i < total; i += stride) {
        float v = acc[i] + bias[i % HC];
        dst[i] = v > 0.f ? v : expm1f(v);
    }
}

// ---------------- global mean pool ----------------
__global__ void pool_sum_kernel(const float* __restrict__ h2, const int* __restrict__ batch,
                                float* __restrict__ gsum, int N, int C) {
    long long i = (long long)blockIdx.x * blockDim.x + threadIdx.x;
    long long total = (long long)N * C;
    long long stride = (long long)gridDim.x * blockDim.x;
    for (; i < total; i += stride) {
        int n = (int)(i / C), c = (int)(i % C);
        atomicAdd(&gsum[(long long)batch[n] * C + c], h2[i]);
    }
}

__global__ void pool_cnt_kernel(const int* __restrict__ batch, float* __restrict__ gcnt, int N) {
    int n = blockIdx.x * blockDim.x + threadIdx.x;
    if (n >= N) return;
    atomicAdd(&gcnt[batch[n]], 1.0f);
}

// ---------------- final linear: out[g, nc] = mean_pool @ lin_w + lin_b ----------------
__global__ void final_kernel(const float* __restrict__ gsum, const float* __restrict__ gcnt,
                             const float* __restrict__ lw, const float* __restrict__ lb,
                             float* __restrict__ out, int G, int C, int NC) {
    int idx = blockIdx.x * blockDim.x + threadIdx.x;
    if (idx >= G * NC) return;
    int g = idx / NC, j = idx % NC;
    float cnt = gcnt[g]; cnt = cnt > 1.f ? cnt : 1.f;
    float a = lb[j];
    for (int c = 0; c < C; ++c) a += (gsum[g * C + c] / cnt) * lw[c * NC + j];
    out[idx] = a;
}

// ---------------- host orchestration ----------------
extern "C" void kernel_launch(void* const* d_in, const int* in_sizes, int n_in,
                              void* d_out, int out_size, void* d_ws, size_t ws_size,
                              hipStream_t stream) {
    const float* x     = (const float*)d_in[0];
    const float* W1    = (const float*)d_in[1];
    const float* asrc1 = (const float*)d_in[2];
    const float* adst1 = (const float*)d_in[3];
    const float* b1    = (const float*)d_in[4];
    const float* W2    = (const float*)d_in[5];
    const float* asrc2 = (const float*)d_in[6];
    const float* adst2 = (const float*)d_in[7];
    const float* b2    = (const float*)d_in[8];
    const float* lin_w = (const float*)d_in[9];
    const float* lin_b = (const float*)d_in[10];
    const int*   ei    = (const int*)d_in[11];
    const int*   batch = (const int*)d_in[12];
    float* out = (float*)d_out;

    const int N    = in_sizes[12];
    const int E    = in_sizes[11] / 2;
    const int ET   = E + N;                 // with self loops
    const int F_IN = in_sizes[0] / N;       // 128
    const int HC1  = in_sizes[4];           // HEADS*HID = 256
    const int HID  = in_sizes[8];           // 64
    const int H1   = HC1 / HID;             // 4
    const int NC   = in_sizes[10];          // 10
    const int G    = out_size / NC;         // 64

    // workspace bump allocator (256B aligned)
    char* base = (char*)d_ws;
    size_t off = 0;
    auto alloc = [&](size_t bytes) -> float* {
        float* p = (float*)(base + off);
        off += (bytes + 255) & ~(size_t)255;
        return p;
    };
    float* bufA  = alloc((size_t)N * HC1 * 4);     // h1, later elu(out1+b1)
    float* bufB  = alloc((size_t)N * HC1 * 4);     // out1 accum; layer2: hp2 + out2
    float* as_s  = alloc((size_t)N * H1 * 4);      // alpha_src (reused layer2)
    float* as_d  = alloc((size_t)N * H1 * 4);
    float* nmax  = alloc((size_t)N * H1 * 4);
    float* nsum  = alloc((size_t)N * H1 * 4);
    float* ew    = alloc((size_t)ET * H1 * 4);     // per-edge softmax weights (reused)
    float* gsum  = alloc((size_t)G * HID * 4);
    float* gcnt  = alloc((size_t)G * 4);
    (void)ws_size;

    float* hp2  = bufB;                       // [N, HID]
    float* out2 = bufB + (size_t)N * HID;     // [N, HID]

    const int TB = 256;
    const float NEG_INF = -3.402823466e38f;   // -FLT_MAX sentinel for segment max

    // ---------------- Layer 1 ----------------
    fill_kernel<<<2048, TB, 0, stream>>>(bufB, (long long)N * HC1, 0.0f);
    fill_kernel<<<256, TB, 0, stream>>>(nmax, (long long)N * H1, NEG_INF);
    fill_kernel<<<256, TB, 0, stream>>>(nsum, (long long)N * H1, 0.0f);

    {   // h1 = x @ W1  (WMMA f32)
        int tiles = (N >> 4) * (HC1 >> 4);
        int blocks = (tiles + (TB >> 5) - 1) / (TB >> 5);
        wmma_gemm_f32<<<blocks, TB, 0, stream>>>(x, W1, bufA, N, F_IN, HC1);
    }
    alpha_kernel<<<(N * H1 + TB - 1) / TB, TB, 0, stream>>>(bufA, asrc1, adst1,
                                                            as_s, as_d, N, H1, HID);
    edge_max_kernel<<<(ET + TB - 1) / TB, TB, 0, stream>>>(ei, as_s, as_d, nmax, E, ET, H1);
    edge_expsum_kernel<<<(ET + TB - 1) / TB, TB, 0, stream>>>(ei, as_s, as_d, nmax, nsum, ew,
                                                              E, ET, H1);
    {   // out1 += alpha * h1[src]
        int edgesPerBlock = TB >> 5;
        int blocks = (ET + edgesPerBlock - 1) / edgesPerBlock;
        edge_scatter_kernel<<<blocks, TB, 0, stream>>>(ei, bufA, ew, nsum, bufB, E, ET, H1, HID);
    }
    // h = elu(out1 + b1) -> bufA
    bias_elu_kernel<<<4096, TB, 0, stream>>>(bufB, b1, bufA, (long long)N * HC1, HC1);

    // ---------------- Layer 2 (1 head) ----------------
    fill_kernel<<<1024, TB, 0, stream>>>(hp2, (long long)N * HID * 2, 0.0f); // zero hp2+out2
    fill_kernel<<<256, TB, 0, stream>>>(nmax, (long long)N, NEG_INF);
    fill_kernel<<<256, TB, 0, stream>>>(nsum, (long long)N, 0.0f);

    {   // hp2 = h @ W2  (WMMA f32)
        int tiles = (N >> 4) * (HID >> 4);
        int blocks = (tiles + (TB >> 5) - 1) / (TB >> 5);
        wmma_gemm_f32<<<blocks, TB, 0, stream>>>(bufA, W2, hp2, N, HC1, HID);
    }
    alpha_kernel<<<(N + TB - 1) / TB, TB, 0, stream>>>(hp2, asrc2, adst2,
                                                       as_s, as_d, N, 1, HID);
    edge_max_kernel<<<(ET + TB - 1) / TB, TB, 0, stream>>>(ei, as_s, as_d, nmax, E, ET, 1);
    edge_expsum_kernel<<<(ET + TB - 1) / TB, TB, 0, stream>>>(ei, as_s, as_d, nmax, nsum, ew,
                                                              E, ET, 1);
    {
        int edgesPerBlock = TB >> 5;
        int blocks = (ET + edgesPerBlock - 1) / edgesPerBlock;
        edge_scatter_kernel<<<blocks, TB, 0, stream>>>(ei, hp2, ew, nsum, out2, E, ET, 1, HID);
    }
    // h2 = elu(out2 + b2) -> reuse hp2 region
    bias_elu_kernel<<<2048, TB, 0, stream>>>(out2, b2, hp2, (long long)N * HID, HID);

    // ---------------- mean pool + final linear ----------------
    fill_kernel<<<1, TB, 0, stream>>>(gsum, (long long)G * HID, 0.0f);
    fill_kernel<<<1, TB, 0, stream>>>(gcnt, (long long)G, 0.0f);
    pool_sum_kernel<<<4096, TB, 0, stream>>>(hp2, batch, gsum, N, HID);
    pool_cnt_kernel<<<(N + TB - 1) / TB, TB, 0, stream>>>(batch, gcnt, N);
    final_kernel<<<(G * NC + TB - 1) / TB, TB, 0, stream>>>(gsum, gcnt, lin_w, lin_b,
                                                            out, G, HID, NC);
}